// SigmaMambaCore_29678224016190
// MI455X (gfx1250) — compile-verified
//
#include <hip/hip_runtime.h>
#include <hip/hip_bf16.h>

#define D_MODEL 768
#define RANK    4
#define BATCH   2
#define SEQ     2048
#define M_ROWS  (BATCH * SEQ)        // 4096
#define N_IN    (D_MODEL * RANK * 2) // 6144

#define BM 64
#define BN 64
#define BK 32

typedef float v2f __attribute__((ext_vector_type(2)));
typedef float v8f __attribute__((ext_vector_type(8)));
typedef int   v4i __attribute__((ext_vector_type(4)));

// ---------------------------------------------------------------------------
// Async global -> LDS copy of 16 bytes per lane (GLOBAL_LOAD_ASYNC_TO_LDS_B128,
// tracked by ASYNCcnt).
// ---------------------------------------------------------------------------
__device__ __forceinline__ void async_copy16(const float* g, float* l) {
#if __has_builtin(__builtin_amdgcn_global_load_async_to_lds_b128)
    __builtin_amdgcn_global_load_async_to_lds_b128(
        (__attribute__((address_space(1))) v4i*)(void*)g,
        (__attribute__((address_space(3))) v4i*)(void*)l,
        /*offset=*/0, /*cpol=*/0);
#else
    float4 v = *(const float4*)g;
    *(float4*)l = v;
#endif
}

__device__ __forceinline__ void async_wait_all() {
#if __has_builtin(__builtin_amdgcn_s_wait_asynccnt)
    __builtin_amdgcn_s_wait_asynccnt(0);
#elif __has_builtin(__builtin_amdgcn_global_load_async_to_lds_b128)
    asm volatile("s_wait_asynccnt 0x0" ::: "memory");
#endif
}

// ---------------------------------------------------------------------------
// RMSNorm: one block per row of x [M_ROWS, D_MODEL]
// ---------------------------------------------------------------------------
__global__ void rmsnorm_kernel(const float* __restrict__ x,
                               const float* __restrict__ w,
                               float* __restrict__ xn) {
    const int row = blockIdx.x;
    const int tid = threadIdx.x;
    const float* xr  = x  + (size_t)row * D_MODEL;
    float*      outr = xn + (size_t)row * D_MODEL;

    float ss = 0.f;
    for (int i = tid; i < D_MODEL; i += 256) {
        float v = xr[i];
        ss += v * v;
    }
    #pragma unroll
    for (int off = 16; off > 0; off >>= 1)
        ss += __shfl_down(ss, off, 32);

    __shared__ float wsum[8];
    if ((tid & 31) == 0) wsum[tid >> 5] = ss;
    __syncthreads();

    float total = 0.f;
    #pragma unroll
    for (int i = 0; i < 8; ++i) total += wsum[i];

    const float scale = rsqrtf(total * (1.0f / D_MODEL) + 1e-6f);
    for (int i = tid; i < D_MODEL; i += 256) {
        outr[i] = xr[i] * scale * w[i];
    }
}

// ---------------------------------------------------------------------------
// Double-buffered LDS-staged WMMA GEMM:  C[M,N] = A[M,K] * B[N,K]^T
//
// Block tile BM=64 x BN=64, K-stage BK=32, 256 threads = 8 wave32.
// Wave (wy,wx): rows [wy*32, wy*32+32), cols [wx*16, wx*16+16)
//   -> 2 accumulators, 16 WMMAs per stage per wave.
//
// Ping-pong LDS buffers: stage i+1 async-loads (ASYNCcnt) are issued right
// after the stage-i barrier and drain while stage i computes; the
// s_wait_asynccnt at the next iteration top hides global latency.
//
// Fragment layouts (ISA 7.12.2, 32-bit data, wave32):
//   A 16x4 : lane L -> M = L%16 ; VGPR j holds K = (L/16)*2 + j
//   B 4x16 : lane L -> N = L%16 ; VGPR j holds K = (L/16)*2 + j
//   C 16x16: lane L -> N = L%16 ; VGPR j holds M = j + 8*(L/16)
// ---------------------------------------------------------------------------
__global__ __launch_bounds__(256)
void wmma_gemm_nt(const float* __restrict__ A,   // [M,K] row-major
                  const float* __restrict__ B,   // [N,K] row-major
                  float* __restrict__ C,         // [M,N] row-major
                  int M, int N, int K) {
    __shared__ float lds_a[2][BM * BK];   // 2 x 8 KB
    __shared__ float lds_b[2][BN * BK];   // 2 x 8 KB

    const int tid  = threadIdx.x;
    const int lane = tid & 31;
    const int wave = tid >> 5;         // 0..7
    const int wx   = wave & 3;         // N subtile (16 cols)
    const int wy   = wave >> 2;        // 0..1 (32 rows each)
    const int blockM = blockIdx.y * BM;
    const int blockN = blockIdx.x * BN;

    // stage-copy assignment: each thread moves 2x16B of A and 2x16B of B.
    // Tile is BM x BK = 64x32 floats = 8 KB; rows 0..31 chunk0, 32..63 chunk1.
    const int r0 = tid >> 3;            // 0..31
    const int c0 = (tid & 7) << 2;      // 0,4,...,28
    const float* gA0 = A + (size_t)(blockM + r0)      * K + c0;
    const float* gA1 = A + (size_t)(blockM + r0 + 32) * K + c0;
    const float* gB0 = B + (size_t)(blockN + r0)      * K + c0;
    const float* gB1 = B + (size_t)(blockN + r0 + 32) * K + c0;
    const int lo0 = r0 * BK + c0;
    const int lo1 = (r0 + 32) * BK + c0;

    const int mn = lane & 15;            // M index (A) / N index (B,C)
    const int kk = (lane >> 4) << 1;     // 0 or 2

    v8f acc0 = {};
    v8f acc1 = {};

    const int nstages = K / BK;

    // prologue: stage 0 into buffer 0
    async_copy16(gA0, &lds_a[0][lo0]);
    async_copy16(gA1, &lds_a[0][lo1]);
    async_copy16(gB0, &lds_b[0][lo0]);
    async_copy16(gB1, &lds_b[0][lo1]);

    for (int i = 0; i < nstages; ++i) {
        async_wait_all();       // stage-i data (issued last iteration) arrived
        __syncthreads();        // all waves' stage-i data visible; all waves
                                // done reading buffer (i+1)&1 from stage i-1

        if (i + 1 < nstages) {  // issue stage i+1 into the other buffer
            const int k1  = (i + 1) * BK;
            const int nb  = (i + 1) & 1;
            async_copy16(gA0 + k1, &lds_a[nb][lo0]);
            async_copy16(gA1 + k1, &lds_a[nb][lo1]);
            async_copy16(gB0 + k1, &lds_b[nb][lo0]);
            async_copy16(gB1 + k1, &lds_b[nb][lo1]);
        }

        const int cur = i & 1;
        const float* aBase0 = &lds_a[cur][(wy * 32 + mn) * BK + kk];
        const float* aBase1 = aBase0 + 16 * BK;
        const float* bBase  = &lds_b[cur][(wx * 16 + mn) * BK + kk];

        #pragma unroll
        for (int k = 0; k < BK; k += 4) {
            v2f bf = *(const v2f*)(bBase + k);
            v2f a0 = *(const v2f*)(aBase0 + k);
            v2f a1 = *(const v2f*)(aBase1 + k);
            acc0 = __builtin_amdgcn_wmma_f32_16x16x4_f32(
                       false, a0, false, bf, (short)0, acc0, false, false);
            acc1 = __builtin_amdgcn_wmma_f32_16x16x4_f32(
                       false, a1, false, bf, (short)0, acc1, false, false);
        }
    }

    const int col  = blockN + wx * 16 + mn;
    const int row0 = blockM + wy * 32 + ((lane >> 4) << 3);   // +0 or +8
    #pragma unroll
    for (int j = 0; j < 8; ++j) {
        C[(size_t)(row0 + j)      * N + col] = acc0[j];
        C[(size_t)(row0 + 16 + j) * N + col] = acc1[j];
    }
}

// ---------------------------------------------------------------------------
// Complex linear recurrence  s_t = a*s_{t-1} + b*u_t ;  y = sum_r Re(c*s)
// One thread per (b, d); R=4 recurrences kept in registers.
// ---------------------------------------------------------------------------
__global__ void scan_kernel(const float* __restrict__ u,
                            const float* __restrict__ a_param,
                            const float* __restrict__ b_param,
                            const float* __restrict__ c_param,
                            float* __restrict__ y) {
    const int idx = blockIdx.x * blockDim.x + threadIdx.x;
    if (idx >= BATCH * D_MODEL) return;
    const int b = idx / D_MODEL;
    const int d = idx % D_MODEL;

    float ar[RANK], ai[RANK], br[RANK], bi[RANK], cr[RANK], ci[RANK];
    #pragma unroll
    for (int r = 0; r < RANK; ++r) {
        const int base = (d * RANK + r) * 2;
        ar[r] = tanhf(a_param[base    ]) * 0.97f;
        ai[r] = tanhf(a_param[base + 1]) * 0.97f;
        br[r] = tanhf(b_param[base    ]);
        bi[r] = tanhf(b_param[base + 1]);
        cr[r] = tanhf(c_param[base    ]);
        ci[r] = tanhf(c_param[base + 1]);
    }

    float sr[RANK] = {0.f, 0.f, 0.f, 0.f};
    float si[RANK] = {0.f, 0.f, 0.f, 0.f};

    for (int t = 0; t < SEQ; ++t) {
        const size_t ubase =
            (((size_t)(b * SEQ + t) * D_MODEL) + d) * (RANK * 2);
        float acc = 0.f;
        #pragma unroll
        for (int r = 0; r < RANK; ++r) {
            const float ur = u[ubase + 2 * r];
            const float ui = u[ubase + 2 * r + 1];
            const float bur = br[r] * ur - bi[r] * ui;
            const float bui = br[r] * ui + bi[r] * ur;
            const float nsr = ar[r] * sr[r] - ai[r] * si[r] + bur;
            const float nsi = ar[r] * si[r] + ai[r] * sr[r] + bui;
            sr[r] = nsr;
            si[r] = nsi;
            acc += cr[r] * nsr - ci[r] * nsi;
        }
        y[(size_t)(b * SEQ + t) * D_MODEL + d] = acc;
    }
}

// ---------------------------------------------------------------------------
// Launch
// ---------------------------------------------------------------------------
extern "C" void kernel_launch(void* const* d_in, const int* in_sizes, int n_in,
                              void* d_out, int out_size, void* d_ws, size_t ws_size,
                              hipStream_t stream) {
    const float* x       = (const float*)d_in[0];
    const float* norm_w  = (const float*)d_in[1];
    const float* W_in    = (const float*)d_in[2];  // [N_IN, D_MODEL]
    const float* W_out   = (const float*)d_in[3];  // [D_MODEL, D_MODEL]
    const float* a_param = (const float*)d_in[4];
    const float* b_param = (const float*)d_in[5];
    const float* c_param = (const float*)d_in[6];
    float* out = (float*)d_out;

    // workspace layout
    float* xn = (float*)d_ws;                         // [M_ROWS, D_MODEL]
    float* u  = xn + (size_t)M_ROWS * D_MODEL;        // [M_ROWS, N_IN]
    float* y  = u  + (size_t)M_ROWS * N_IN;           // [M_ROWS, D_MODEL]

    // 1) RMSNorm
    rmsnorm_kernel<<<M_ROWS, 256, 0, stream>>>(x, norm_w, xn);

    // 2) u = xn @ W_in^T   (M=4096, N=6144, K=768)
    {
        dim3 block(256);
        dim3 grid(N_IN / BN, M_ROWS / BM);            // (96, 64)
        wmma_gemm_nt<<<grid, block, 0, stream>>>(xn, W_in, u,
                                                 M_ROWS, N_IN, D_MODEL);
    }

    // 3) complex recurrence + output mix -> y
    scan_kernel<<<(BATCH * D_MODEL + 255) / 256, 256, 0, stream>>>(
        u, a_param, b_param, c_param, y);

    // 4) out = y @ W_out^T  (M=4096, N=768, K=768)
    {
        dim3 block(256);
        dim3 grid(D_MODEL / BN, M_ROWS / BM);         // (12, 64)
        wmma_gemm_nt<<<grid, block, 0, stream>>>(y, W_out, out,
                                                 M_ROWS, D_MODEL, D_MODEL);
    }

    (void)in_sizes; (void)n_in; (void)out_size; (void)ws_size;
}